// AutoregressiveFlow_13280038880145
// MI455X (gfx1250) — compile-verified
//
#include <hip/hip_runtime.h>
#include <hip/hip_bf16.h>
#include <math.h>

// MI455X / gfx1250, wave32. fp32 WMMA (16x16x4) for exact-precision GRU GEMMs.

typedef float v2f __attribute__((ext_vector_type(2)));
typedef float v4f __attribute__((ext_vector_type(4)));
typedef float v8f __attribute__((ext_vector_type(8)));

#define WAVES_PER_BLOCK 8
#define BLOCK_THREADS   (WAVES_PER_BLOCK * 32)
#define T_STEPS 12
#define HDIM    64
#define EPSV    1e-6f
#define LOG2E   1.4426950408889634f

// B-operand slots in LDS (each slot = 64 floats, WMMA-B register order):
//   [0,204)   : gh GEMM, tile j (0..11 over 192 output cols), chunk c (0..16); c==0 is the
//               augmented chunk (Wih cols + fused bih+bhh for r/z tiles, bhh for n tiles)
//   [204,208) : gi_n (input path of n-gate), one aug chunk per n tile
//   [208,242) : MLP1 (W1.T, 64->32), tile j (0..1), chunk c (0..16); c==0 carries b1
//   [242,251) : MLP2 (W2.T, 32->4 padded to 16), chunk c (0..8); c==0 carries b2
#define NOPS      251
#define WB_FLOATS (NOPS * 64)
#define PW_FLOATS 1632   // per-wave: h 16x64 + s 16x32 + y 16x2(pad 32) + dls 16x4

// ---- fast hardware transcendentals (TRANS unit: v_exp_f32/v_tanh_f32/v_rcp_f32) ----
__device__ __forceinline__ float fast_exp2(float x) {
#if __has_builtin(__builtin_amdgcn_exp2f)
  return __builtin_amdgcn_exp2f(x);
#else
  return exp2f(x);
#endif
}
__device__ __forceinline__ float fast_rcp(float x) {
#if __has_builtin(__builtin_amdgcn_rcpf)
  return __builtin_amdgcn_rcpf(x);
#else
  return 1.0f / x;
#endif
}
__device__ __forceinline__ float fast_tanh(float x) {
#if __has_builtin(__builtin_amdgcn_tanhf)
  return __builtin_amdgcn_tanhf(x);
#elif __has_builtin(__builtin_amdgcn_tanh_f32)
  return __builtin_amdgcn_tanh_f32(x);
#else
  // tanh(x) = 1 - 2/(exp2(2x*log2e) + 1)
  float e = fast_exp2(2.0f * LOG2E * x);
  return 1.0f - 2.0f * fast_rcp(e + 1.0f);
#endif
}
__device__ __forceinline__ float sigmoidf_(float x) {
  return fast_rcp(1.0f + fast_exp2(-LOG2E * x));
}
__device__ __forceinline__ float softplusf_(float x) {
  return (x > 20.0f) ? x : log1pf(expf(x));   // accurate path: feeds output scale
}

__device__ __forceinline__ v8f wmma_f32(v2f a, v2f b, v8f c) {
  // 8 args: (neg_a, A, neg_b, B, c_mod, C, reuse_a, reuse_b)
  return __builtin_amdgcn_wmma_f32_16x16x4_f32(false, a, false, b, (short)0, c, false, false);
}

__global__ __launch_bounds__(BLOCK_THREADS)
void arflow_gru_wmma(const float* __restrict__ y_tm1, const float* __restrict__ z,
                     const float* __restrict__ x,     const float* __restrict__ Wih,
                     const float* __restrict__ Whh,   const float* __restrict__ bih,
                     const float* __restrict__ bhh,   const float* __restrict__ W1,
                     const float* __restrict__ b1,    const float* __restrict__ W2,
                     const float* __restrict__ b2,    float* __restrict__ out, int B) {
  extern __shared__ float smem[];
  float* wB = smem;
  const int tid = threadIdx.x;

  // ---- stage all B operands (weights, biases fused) into LDS, operand order ----
  for (int idx = tid; idx < WB_FLOATS; idx += BLOCK_THREADS) {
    int op = idx >> 6;
    int r  = idx & 63;
    int l  = r >> 1;                 // lane this float feeds
    int p  = r & 1;                  // which of the 2 B VGPRs
    int kl = ((l >> 4) << 1) + p;    // k within 4-chunk: lanes 0-15 -> K{0,1}, 16-31 -> K{2,3}
    int nn = l & 15;                 // column within 16-wide N tile
    float val = 0.0f;
    if (op < 204) {
      int j = op / 17, c = op % 17;
      int n = j * 16 + nn;           // output column 0..191
      if (c == 0) {                  // aug chunk, A row = [y0, y1, 1, 0]
        if      (kl == 0) val = (j < 8) ? Wih[n * 2 + 0] : 0.0f;
        else if (kl == 1) val = (j < 8) ? Wih[n * 2 + 1] : 0.0f;
        else if (kl == 2) val = (j < 8) ? (bih[n] + bhh[n]) : bhh[n];
      } else {
        int k = (c - 1) * 4 + kl;    // 0..63
        val = Whh[n * HDIM + k];     // Whh.T[k, n]
      }
    } else if (op < 208) {           // gi_n = y @ Wih_n.T + bih_n
      int n = 128 + (op - 204) * 16 + nn;
      if      (kl == 0) val = Wih[n * 2 + 0];
      else if (kl == 1) val = Wih[n * 2 + 1];
      else if (kl == 2) val = bih[n];
    } else if (op < 242) {           // MLP1
      int o = op - 208;
      int j = o / 17, c = o % 17;
      int n = j * 16 + nn;           // 0..31
      if (c == 0) { if (kl == 0) val = b1[n]; }   // A row = [1,0,0,0]
      else        { int k = (c - 1) * 4 + kl; val = W1[n * HDIM + k]; }
    } else {                         // MLP2 (N padded 4->16)
      int c = op - 242;
      if (nn < 4) {
        if (c == 0) { if (kl == 0) val = b2[nn]; }
        else        { int k = (c - 1) * 4 + kl; val = W2[nn * 32 + k]; }
      }
    }
    wB[idx] = val;
  }
  __syncthreads();

  const int wave = tid >> 5;
  const int lane = tid & 31;
  const int lo   = lane & 15;
  const int hi   = lane >> 4;        // half-wave select
  const int tile = blockIdx.x * WAVES_PER_BLOCK + wave;
  if (tile * 16 >= B) return;        // wave-uniform: EXEC stays all-ones for WMMA
  const int b0 = tile * 16;

  float* pw     = smem + WB_FLOATS + wave * PW_FLOATS;
  float* hBuf   = pw;                // 16 x 64 row-major (A-operand source)
  float* sBuf   = pw + 1024;         // 16 x 32 row-major
  float* yBuf   = pw + 1536;         // 16 x 2
  float* dlsBuf = pw + 1568;         // 16 x 4

  // h0 = z : load straight into D-layout regs + LDS A-source tile
  v8f hD[4];
  #pragma unroll
  for (int g = 0; g < 4; ++g) {
    #pragma unroll
    for (int v = 0; v < 8; ++v) {
      int m = v + 8 * hi, col = 16 * g + lo;
      float hv = z[(size_t)(b0 + m) * HDIM + col];
      hD[g][v] = hv;
      hBuf[m * HDIM + col] = hv;
    }
  }
  if (lane < 16) {
    v2f yv = *(const v2f*)(y_tm1 + (size_t)(b0 + lane) * 2);
    *(v2f*)(yBuf + lane * 2) = yv;
  }

  #pragma unroll 1
  for (int t = 0; t < T_STEPS; ++t) {
    // augmented A chunk: rows -> [y0, y1, 1, 0]
    v2f yv = *(const v2f*)(yBuf + lo * 2);
    v2f a0; a0.x = hi ? 1.0f : yv.x; a0.y = hi ? 0.0f : yv.y;
    v2f a1; a1.x = hi ? 0.0f : 1.0f; a1.y = 0.0f;   // [1,0,0,0] for bias rows

    // ---- GRU gates, one 16-column slice (g) at a time ----
    v8f hNew[4];
    #pragma unroll
    for (int g = 0; g < 4; ++g) {
      v8f accR = {}, accZ = {}, accN = {}, giN = {};
      {
        v2f bR = *(const v2f*)(wB + ((g    ) * 17) * 64 + lane * 2);
        v2f bZ = *(const v2f*)(wB + ((g + 4) * 17) * 64 + lane * 2);
        v2f bN = *(const v2f*)(wB + ((g + 8) * 17) * 64 + lane * 2);
        accR = wmma_f32(a0, bR, accR);
        accZ = wmma_f32(a0, bZ, accZ);
        accN = wmma_f32(a0, bN, accN);
      }
      #pragma unroll
      for (int c = 1; c <= 16; ++c) {
        v2f ap = *(const v2f*)(hBuf + lo * HDIM + (c - 1) * 4 + 2 * hi);
        v2f bR = *(const v2f*)(wB + ((g    ) * 17 + c) * 64 + lane * 2);
        v2f bZ = *(const v2f*)(wB + ((g + 4) * 17 + c) * 64 + lane * 2);
        v2f bN = *(const v2f*)(wB + ((g + 8) * 17 + c) * 64 + lane * 2);
        accR = wmma_f32(ap, bR, accR);
        accZ = wmma_f32(ap, bZ, accZ);
        accN = wmma_f32(ap, bN, accN);
      }
      {
        v2f bG = *(const v2f*)(wB + (204 + g) * 64 + lane * 2);
        giN = wmma_f32(a0, bG, giN);
      }
      #pragma unroll
      for (int v = 0; v < 8; ++v) {
        float r = sigmoidf_(accR[v]);
        float u = sigmoidf_(accZ[v]);
        float n = fast_tanh(giN[v] + r * accN[v]);
        hNew[g][v] = (1.0f - u) * n + u * hD[g][v];
      }
    }
    // commit h (all slices after every gate GEMM consumed old h)
    #pragma unroll
    for (int g = 0; g < 4; ++g) {
      hD[g] = hNew[g];
      #pragma unroll
      for (int v = 0; v < 8; ++v)
        hBuf[(v + 8 * hi) * HDIM + 16 * g + lo] = hNew[g][v];
    }

    // ---- MLP head: relu(h W1.T + b1) W2.T + b2 ----
    #pragma unroll
    for (int jt = 0; jt < 2; ++jt) {
      v8f acc = {};
      { v2f bb = *(const v2f*)(wB + (208 + jt * 17) * 64 + lane * 2);
        acc = wmma_f32(a1, bb, acc); }
      #pragma unroll
      for (int c = 1; c <= 16; ++c) {
        v2f ap = *(const v2f*)(hBuf + lo * HDIM + (c - 1) * 4 + 2 * hi);
        v2f bb = *(const v2f*)(wB + (208 + jt * 17 + c) * 64 + lane * 2);
        acc = wmma_f32(ap, bb, acc);
      }
      #pragma unroll
      for (int v = 0; v < 8; ++v)
        sBuf[(v + 8 * hi) * 32 + 16 * jt + lo] = fmaxf(acc[v], 0.0f);
    }
    v8f dacc = {};
    { v2f bb = *(const v2f*)(wB + 242 * 64 + lane * 2);
      dacc = wmma_f32(a1, bb, dacc); }
    #pragma unroll
    for (int c = 1; c <= 8; ++c) {
      v2f ap = *(const v2f*)(sBuf + lo * 32 + (c - 1) * 4 + 2 * hi);
      v2f bb = *(const v2f*)(wB + (242 + c) * 64 + lane * 2);
      dacc = wmma_f32(ap, bb, dacc);
    }

    // ---- flow update: y_t = y_prev + dloc + (softplus(dls[:,2:]) + eps) * x_t ----
    if (lo < 4) {
      #pragma unroll
      for (int v = 0; v < 8; ++v)
        dlsBuf[(v + 8 * hi) * 4 + lo] = dacc[v];
    }
    if (lane < 16) {
      v4f d  = *(const v4f*)(dlsBuf + lane * 4);
      v2f yp = *(const v2f*)(yBuf + lane * 2);
      size_t xoff = ((size_t)(b0 + lane) * T_STEPS + t) * 2;
      v2f xt = *(const v2f*)(x + xoff);
      float s0 = softplusf_(d.z) + EPSV;
      float s1 = softplusf_(d.w) + EPSV;
      v2f yn;
      yn.x = yp.x + d.x + s0 * xt.x;
      yn.y = yp.y + d.y + s1 * xt.y;
      *(v2f*)(yBuf + lane * 2) = yn;
      *(v2f*)(out + xoff) = yn;
    }
  }
}

extern "C" void kernel_launch(void* const* d_in, const int* in_sizes, int n_in,
                              void* d_out, int out_size, void* d_ws, size_t ws_size,
                              hipStream_t stream) {
  const float* y_tm1 = (const float*)d_in[0];
  const float* z     = (const float*)d_in[1];
  const float* x     = (const float*)d_in[2];
  const float* Wih   = (const float*)d_in[3];
  const float* Whh   = (const float*)d_in[4];
  const float* bih   = (const float*)d_in[5];
  const float* bhh   = (const float*)d_in[6];
  const float* W1    = (const float*)d_in[7];
  const float* b1    = (const float*)d_in[8];
  const float* W2    = (const float*)d_in[9];
  const float* b2    = (const float*)d_in[10];
  float* out = (float*)d_out;

  int B = in_sizes[0] / 2;
  int numTiles = (B + 15) / 16;
  int grid = (numTiles + WAVES_PER_BLOCK - 1) / WAVES_PER_BLOCK;
  size_t shmem = (size_t)(WB_FLOATS + WAVES_PER_BLOCK * PW_FLOATS) * sizeof(float);

  arflow_gru_wmma<<<grid, BLOCK_THREADS, shmem, stream>>>(
      y_tm1, z, x, Wih, Whh, bih, bhh, W1, b1, W2, b2, out, B);
}